// Separator_55791625175550
// MI455X (gfx1250) — compile-verified
//
#include <hip/hip_runtime.h>

// ---------------------------------------------------------------------------
// Multichannel Wiener EM (F=1000, B=2049, C=2, S=4, 2 iterations).
// Memory-bound: ~850 MB total traffic -> ~36 us @ 23.3 TB/s HBM; the 131 MB
// intermediate fits in the 192 MB L2, so iteration 2 is mostly L2-resident.
// The per-bin spatial-covariance reduction is a 16x16 f32 Gram over F, which
// maps exactly onto V_WMMA_F32_16X16X4_F32 (A == B registers for a Gram).
// ---------------------------------------------------------------------------

#define F_DIM 1000
#define B_DIM 2049
#define CELL  16                     // floats per (f,b) cell of y: C*2*S
#define ROWSTRIDE (B_DIM * CELL)     // floats per frame row of y
#define NCHUNK (F_DIM / 4)           // 250 K=4 chunks of frames
#define EPS_F      1.1920928955078125e-07f   // FLT_EPSILON
#define SQRT_EPS_F 3.4526698300141308e-04f   // sqrt(FLT_EPSILON)

typedef __attribute__((ext_vector_type(2))) float v2f;
typedef __attribute__((ext_vector_type(8))) float v8f;

// ---------------------------------------------------------------------------
// Kernel 1: per-bin E-step reduction.
// One block per bin b. Each wave accumulates a 16x16 f32 Gram of the 16
// component-vectors (4 sources x {re,im} x 2 ch) over its share of the 1000
// frames using V_WMMA_F32_16X16X4_F32 (K = 4 frames / instruction).
// A-matrix layout (ISA 7.12.2, 32-bit A 16x4): lanes 0-15 hold K=0(V0),K=1(V1);
// lanes 16-31 hold K=2(V0),K=3(V1); M = lane & 15. For a Gram, B's row-striped
// layout holds the same values -> pass A's registers for both operands.
// The wave id is pushed into an SGPR via readfirstlane so the chunk loop is a
// scalar loop (EXEC provably all-1s for WMMA, no per-lane exec bookkeeping),
// and it is unrolled 2x onto independent accumulators to break the
// WMMA->WMMA RAW hazard chain and keep more loads in flight.
// Cross-wave reduce in LDS, then 4 lanes emit normalized R (B,C,C,2,S).
// ---------------------------------------------------------------------------
__global__ __launch_bounds__(256) void reduce_R(const float* __restrict__ y,
                                                float* __restrict__ Rn) {
  const int b    = blockIdx.x;
  const int tid  = threadIdx.x;
  const int lane = tid & 31;
  const int half = lane >> 4;     // 0: K=0,1   1: K=2,3
  const int m    = lane & 15;     // matrix row (component index)
  // wave index as a true SGPR -> scalar loop control below
  const int wave = __builtin_amdgcn_readfirstlane(tid >> 5);   // 0..7

  __shared__ float partial[8 * 256];
  __shared__ float G[256];        // full Gram, G[M*16+N]

  const float* ybase = y + (size_t)b * CELL + m;

  v8f acc0 = {};                  // two independent 16x16 f32 accumulators
  v8f acc1 = {};
  int ch = wave;
  // pairs (ch, ch+8): independent WMMAs, 4 loads in flight per iteration
  for (; ch + 8 < NCHUNK; ch += 16) {
    const float* p0 = ybase + (size_t)(ch * 4 + 2 * half) * ROWSTRIDE;
    const float* p1 = ybase + (size_t)((ch + 8) * 4 + 2 * half) * ROWSTRIDE;
    v2f a0, a1;
    a0.x = p0[0];                 // frame K = 2*half
    a0.y = p0[ROWSTRIDE];         // frame K = 2*half + 1
    a1.x = p1[0];
    a1.y = p1[ROWSTRIDE];
    acc0 = __builtin_amdgcn_wmma_f32_16x16x4_f32(false, a0, false, a0,
                                                 (short)0, acc0, false, false);
    acc1 = __builtin_amdgcn_wmma_f32_16x16x4_f32(false, a1, false, a1,
                                                 (short)0, acc1, false, false);
  }
  if (ch < NCHUNK) {              // odd leftover chunk (waves 2..7)
    const float* p0 = ybase + (size_t)(ch * 4 + 2 * half) * ROWSTRIDE;
    v2f a0;
    a0.x = p0[0];
    a0.y = p0[ROWSTRIDE];
    acc0 = __builtin_amdgcn_wmma_f32_16x16x4_f32(false, a0, false, a0,
                                                 (short)0, acc0, false, false);
  }

#pragma unroll
  for (int r = 0; r < 8; ++r)
    partial[wave * 256 + r * 32 + lane] = acc0[r] + acc1[r];
  __syncthreads();

  // Sum the 8 per-wave Grams; C/D layout: VGPR r, lane l -> M = r + 8*(l>>4),
  // N = l & 15.
  {
    float g = 0.f;
#pragma unroll
    for (int w = 0; w < 8; ++w) g += partial[w * 256 + tid];
    const int r = tid >> 5, l = tid & 31;
    const int M = r + 8 * (l >> 4);
    const int N = l & 15;
    G[M * 16 + N] = g;
  }
  __syncthreads();

  // Component index map per source s: a0=s, b0=4+s, a1=8+s, b1=12+s.
  if (tid < 4) {
    const int s = tid;
    const float d00  = G[s * 16 + s]           + G[(4 + s) * 16 + (4 + s)];
    const float d11  = G[(8 + s) * 16 + 8 + s] + G[(12 + s) * 16 + 12 + s];
    const float re01 = G[s * 16 + 8 + s]       + G[(4 + s) * 16 + 12 + s];
    const float im01 = G[(4 + s) * 16 + 8 + s] - G[s * 16 + 12 + s];
    const float sumv = 0.5f * (d00 + d11);           // sum_f mean_c |y|^2
    const float nrm  = 1.0f / (EPS_F + sumv);
    float* Rb = Rn + (size_t)b * 32;                 // (b, c, d, {re,im}, s)
    Rb[0  + s] =  d00  * nrm;   // R[0,0] re
    Rb[4  + s] =  0.f;          // R[0,0] im (exactly 0 in reference too)
    Rb[8  + s] =  re01 * nrm;   // R[0,1] re
    Rb[12 + s] =  im01 * nrm;   // R[0,1] im
    Rb[16 + s] =  re01 * nrm;   // R[1,0] = conj(R[0,1])
    Rb[20 + s] = -im01 * nrm;
    Rb[24 + s] =  d11  * nrm;   // R[1,1] re
    Rb[28 + s] =  0.f;
  }
}

// ---------------------------------------------------------------------------
// Kernel 2: M-step. One block per bin b (R cached in LDS/regs), threads
// stride over frames. Per cell: v_s, Cxx = sqrt(eps)I + sum_s v_s R_s,
// 2x2 Hermitian inverse, t = Cxx^-1 x, y_s = v_s * R_s t.
// yin may alias yout (in-place): the full 64B cell is read before writing.
// ---------------------------------------------------------------------------
__global__ __launch_bounds__(256) void apply_wiener(const float* yin,
                                                    const float* __restrict__ x,
                                                    const float* __restrict__ Rn,
                                                    float* yout,
                                                    float* __restrict__ vout,
                                                    int writeV) {
  const int b = blockIdx.x;
  __shared__ float Rl[32];
  if (threadIdx.x < 32) Rl[threadIdx.x] = Rn[(size_t)b * 32 + threadIdx.x];
  __syncthreads();

  float r00[4], r01re[4], r01im[4], r11[4];
#pragma unroll
  for (int s = 0; s < 4; ++s) {
    r00[s]   = Rl[0 + s];
    r01re[s] = Rl[8 + s];
    r01im[s] = Rl[12 + s];
    r11[s]   = Rl[24 + s];
  }

  for (int f = threadIdx.x; f < F_DIM; f += 256) {
    const size_t cell = (size_t)f * B_DIM + b;
    const float* yp = yin + cell * CELL;
    float yv[16];
#pragma unroll
    for (int i = 0; i < 16; ++i) yv[i] = yp[i];   // full cell before any store

    const float* xp = x + cell * 4;
    const float xr0 = xp[0], xi0 = xp[1], xr1 = xp[2], xi1 = xp[3];

    float v[4];
    float c00 = SQRT_EPS_F, c11 = SQRT_EPS_F, c01re = 0.f, c01im = 0.f;
#pragma unroll
    for (int s = 0; s < 4; ++s) {
      const float a0 = yv[s], b0 = yv[4 + s], a1 = yv[8 + s], b1 = yv[12 + s];
      v[s] = 0.5f * (a0 * a0 + b0 * b0 + a1 * a1 + b1 * b1);
      c00   += v[s] * r00[s];
      c01re += v[s] * r01re[s];
      c01im += v[s] * r01im[s];
      c11   += v[s] * r11[s];
    }
    if (writeV) {
#pragma unroll
      for (int s = 0; s < 4; ++s) vout[cell * 4 + s] = v[s];
    }

    // det = c00*c11 - |c01|^2 (imag exactly 0); inv_det = dre / dre^2
    const float dre = c00 * c11 - (c01re * c01re + c01im * c01im);
    const float g   = dre / (dre * dre);
    const float i00 = g * c11, i11 = g * c00;
    const float i01re = -g * c01re, i01im = -g * c01im;   // i10 = conj(i01)

    // t = Cxx^-1 @ x
    const float t0re = i00 * xr0 + i01re * xr1 - i01im * xi1;
    const float t0im = i00 * xi0 + i01re * xi1 + i01im * xr1;
    const float t1re = i01re * xr0 + i01im * xi0 + i11 * xr1;
    const float t1im = i01re * xi0 - i01im * xr0 + i11 * xi1;

    float* op = yout + cell * CELL;
#pragma unroll
    for (int s = 0; s < 4; ++s) {
      op[s]      = v[s] * (r00[s] * t0re + r01re[s] * t1re - r01im[s] * t1im);
      op[4 + s]  = v[s] * (r00[s] * t0im + r01re[s] * t1im + r01im[s] * t1re);
      op[8 + s]  = v[s] * (r01re[s] * t0re + r01im[s] * t0im + r11[s] * t1re);
      op[12 + s] = v[s] * (r01re[s] * t0im - r01im[s] * t0re + r11[s] * t1im);
    }
  }
}

// ---------------------------------------------------------------------------
// Host-side launcher. iterations == 2 (per setup_inputs). Zero workspace:
// iter-1 results live in d_out regions and are overwritten by iter 2.
// ---------------------------------------------------------------------------
extern "C" void kernel_launch(void* const* d_in, const int* in_sizes, int n_in,
                              void* d_out, int out_size, void* d_ws, size_t ws_size,
                              hipStream_t stream) {
  (void)in_sizes; (void)n_in; (void)out_size; (void)d_ws; (void)ws_size;

  const float* y = (const float*)d_in[0];   // (F,B,C,2,S)
  const float* x = (const float*)d_in[1];   // (F,B,C,2)
  // d_in[2] = iterations (== 2)

  float* out   = (float*)d_out;
  float* y_out = out;                                   // F*B*C*2*S = 32,784,000
  float* v_out = out + (size_t)F_DIM * B_DIM * CELL;    // F*B*S     =  8,196,000
  float* R_out = v_out + (size_t)F_DIM * B_DIM * 4;     // B*C*C*2*S =     65,568

  dim3 grid(B_DIM), block(256);

  // iteration 1: E-step from input y, M-step writes filtered y into y_out
  reduce_R<<<grid, block, 0, stream>>>(y, R_out);
  apply_wiener<<<grid, block, 0, stream>>>(y, x, R_out, y_out, nullptr, 0);

  // iteration 2: E-step from iter-1 y (final R, final v), M-step in place
  reduce_R<<<grid, block, 0, stream>>>(y_out, R_out);
  apply_wiener<<<grid, block, 0, stream>>>(y_out, x, R_out, y_out, v_out, 1);
}